// Edge_level_5145370821343
// MI455X (gfx1250) — compile-verified
//
#include <hip/hip_runtime.h>
#include <hip/hip_bf16.h>

// ---------------------------------------------------------------------------
// Edge-level GAT on MI455X (gfx1250, wave32).
// Dense GEMMs via V_WMMA_F32_16X16X4_F32 (full fp32 precision — workload is
// scatter/atomic bound, so low-precision WMMA buys nothing and costs accuracy).
// LDS staging uses GLOBAL_LOAD_ASYNC_TO_LDS_B128 (ASYNCcnt) when available.
// ---------------------------------------------------------------------------

typedef __attribute__((ext_vector_type(2))) float v2f;
typedef __attribute__((ext_vector_type(8))) float v8f;
typedef __attribute__((ext_vector_type(4))) int   v4i;

#define HEADS 4
#define DOUT 16
#define EDIM 5
#define FIN 64
#define NEG_SLOPE 0.2f

// padded LDS row strides (floats) to break bank conflicts:
//  - A tile read walks K with row stride: pad 68 -> bank step 4 per row
//  - W tile read: lane-halves differ by K+2: pad 72 -> bank step 8 per K row
#define SA_STRIDE 68
#define SW_STRIDE 72

// ---------------- async global->LDS staging (guarded) ----------------------
#if __has_builtin(__builtin_amdgcn_global_load_async_to_lds_b128)
#define HAVE_ASYNC_LDS 1
typedef __attribute__((address_space(1))) v4i* gptr_v4i;
typedef __attribute__((address_space(3))) v4i* lptr_v4i;
#endif

__device__ __forceinline__ void cp16_g2l(const float* g, float* l) {
#if defined(HAVE_ASYNC_LDS)
    __builtin_amdgcn_global_load_async_to_lds_b128(
        (gptr_v4i)g, (lptr_v4i)l, 0, 0);
#else
    *(float4*)l = *(const float4*)g;
#endif
}

__device__ __forceinline__ void staging_fence() {
#if defined(HAVE_ASYNC_LDS)
#if __has_builtin(__builtin_amdgcn_s_wait_asynccnt)
    __builtin_amdgcn_s_wait_asynccnt(0);
#else
    asm volatile("s_wait_asynccnt 0x0" ::: "memory");
#endif
#endif
    __syncthreads();
}

// ---- monotone float<->uint mapping for atomicMax on floats ----------------
__device__ __forceinline__ unsigned fmap(float f) {
    unsigned u = __float_as_uint(f);
    return (u & 0x80000000u) ? ~u : (u | 0x80000000u);
}
__device__ __forceinline__ float funmap(unsigned u) {
    u = (u & 0x80000000u) ? (u & 0x7FFFFFFFu) : ~u;
    return __uint_as_float(u);
}

// ---------------------------------------------------------------------------
__global__ void k_zero(float* __restrict__ p, int n) {
    int i = blockIdx.x * blockDim.x + threadIdx.x;
    if (i < n) p[i] = 0.0f;
}

// ---------------------------------------------------------------------------
// Kernel 1: feat_p = feat @ W_fc   ([N,64] @ [64,64] -> [N,64]) via fp32 WMMA.
// 128 threads = 4 waves; wave w owns rows [blk*64 + w*16, +16).
// Both operands staged in LDS (async when available); inner loop is
// ds_load(A pair) + 4 x { ds_load(B pair) + wmma } with 4 accumulators.
// A layout (16x4 f32): M = lane%16, lane-half selects K pair {0,1}/{2,3}.
// B layout: N = lane%16, same K-pair rule. C/D: M = reg + 8*half, N = lane%16.
// ---------------------------------------------------------------------------
__global__ __launch_bounds__(128)
void k_node_proj(const float* __restrict__ feat, const float* __restrict__ Wfc,
                 float* __restrict__ featp, int N) {
    __shared__ float sW[FIN * SW_STRIDE];   // 18.0 KB
    __shared__ float sA[FIN * SA_STRIDE];   // 17.0 KB

    // stage W_fc (64x64) and this block's 64-row feat tile, 16B chunks
    for (int j = threadIdx.x; j < 64 * 16; j += 128) {
        int r = j >> 4, c4 = (j & 15) << 2;
        cp16_g2l(&Wfc[r * FIN + c4], &sW[r * SW_STRIDE + c4]);
    }
    for (int j = threadIdx.x; j < 64 * 16; j += 128) {
        int r = j >> 4, c4 = (j & 15) << 2;
        int grow = blockIdx.x * 64 + r;
        if (grow >= N) grow = N - 1;           // clamp: dead rows never stored
        cp16_g2l(&feat[(size_t)grow * FIN + c4], &sA[r * SA_STRIDE + c4]);
    }
    staging_fence();

    const int lane = threadIdx.x & 31;
    const int wave = threadIdx.x >> 5;
    const int n16  = lane & 15;
    const int half = lane >> 4;
    const int kbase = half * 2;
    const int rowBase = blockIdx.x * 64 + wave * 16;
    const int lr = wave * 16 + n16;            // local A row for this lane

    v8f acc[4] = {};
    for (int kk = 0; kk < FIN; kk += 4) {
        const int k0 = kk + kbase;
        v2f a;
        a.x = sA[lr * SA_STRIDE + k0];
        a.y = sA[lr * SA_STRIDE + k0 + 1];
        #pragma unroll
        for (int t = 0; t < 4; ++t) {
            v2f b;
            b.x = sW[k0 * SW_STRIDE + t * 16 + n16];
            b.y = sW[(k0 + 1) * SW_STRIDE + t * 16 + n16];
            acc[t] = __builtin_amdgcn_wmma_f32_16x16x4_f32(
                false, a, false, b, (short)0, acc[t], false, false);
        }
    }

    // epilogue: uniform fast path (all 16 wave rows in-bounds) with
    // immediate-offset stores; guarded path only for the single tail block.
    if (rowBase + 16 <= N) {
        #pragma unroll
        for (int t = 0; t < 4; ++t) {
            float* outp = featp + (size_t)(rowBase + half * 8) * FIN + t * 16 + n16;
            #pragma unroll
            for (int r = 0; r < 8; ++r)
                outp[(size_t)r * FIN] = acc[t][r];
        }
    } else {
        #pragma unroll
        for (int t = 0; t < 4; ++t) {
            #pragma unroll
            for (int r = 0; r < 8; ++r) {
                int grow = rowBase + r + half * 8;  // C/D layout: M = r + 8*half
                if (grow < N)
                    featp[(size_t)grow * FIN + t * 16 + n16] = acc[t][r];
            }
        }
    }
}

// ---------------------------------------------------------------------------
// Kernel 2: el[n,h], er[n,h]
// ---------------------------------------------------------------------------
__global__ void k_node_attn(const float* __restrict__ featp,
                            const float* __restrict__ attn_l,
                            const float* __restrict__ attn_r,
                            float* __restrict__ el, float* __restrict__ er, int NH) {
    int t = blockIdx.x * blockDim.x + threadIdx.x;
    if (t >= NH) return;
    int n = t >> 2, h = t & 3;
    const float* fp = featp + (size_t)n * FIN + h * DOUT;
    float sl = 0.0f, sr = 0.0f;
    #pragma unroll
    for (int d = 0; d < DOUT; ++d) {
        float v = fp[d];
        sl += v * attn_l[h * DOUT + d];
        sr += v * attn_r[h * DOUT + d];
    }
    el[t] = sl; er[t] = sr;
}

// ---------------------------------------------------------------------------
// Kernel 3: per (edge, head) leaky-relu logit + segment max (atomicMax on
// monotone-mapped uint). edge_p recomputed (25 MACs) instead of stored (128MB).
// ---------------------------------------------------------------------------
__global__ void k_edge_logit(const float* __restrict__ edge_fea,
                             const int* __restrict__ src, const int* __restrict__ dst,
                             const float* __restrict__ Wedg, const float* __restrict__ bedg,
                             const float* __restrict__ attn_edg,
                             const float* __restrict__ el, const float* __restrict__ er,
                             float* __restrict__ elog, unsigned* __restrict__ mhat, int EH) {
    int t = blockIdx.x * blockDim.x + threadIdx.x;
    if (t >= EH) return;
    int e = t >> 2, h = t & 3;
    float ef[EDIM];
    const float* efp = edge_fea + (size_t)e * EDIM;
    #pragma unroll
    for (int i = 0; i < EDIM; ++i) ef[i] = efp[i];
    float ee = 0.0f;
    #pragma unroll
    for (int j = 0; j < EDIM; ++j) {
        float p = bedg[h * EDIM + j];
        #pragma unroll
        for (int i = 0; i < EDIM; ++i)
            p += ef[i] * Wedg[i * (HEADS * EDIM) + h * EDIM + j];
        ee += p * attn_edg[h * EDIM + j];
    }
    int s = src[e], d = dst[e];
    float lv = el[s * HEADS + h] + er[d * HEADS + h] + ee;
    lv = (lv > 0.0f) ? lv : NEG_SLOPE * lv;
    elog[t] = lv;
    atomicMax(mhat + (size_t)d * HEADS + h, fmap(lv));
}

// ---------------------------------------------------------------------------
// Kernel 4: per (edge, head): p = exp(e - m[dst]); unnormalized segment sums:
//   z[dst,h] += p;  ft_f[dst,h,:] += p*edge_p;  ft[dst,h,:] += p*feat_p[src,h,:]
// ---------------------------------------------------------------------------
__global__ void k_edge_aggr(const float* __restrict__ edge_fea,
                            const int* __restrict__ src, const int* __restrict__ dst,
                            const float* __restrict__ Wedg, const float* __restrict__ bedg,
                            const float* __restrict__ featp,
                            const float* __restrict__ elog,
                            const unsigned* __restrict__ mhat,
                            float* __restrict__ z, float* __restrict__ ftf,
                            float* __restrict__ ft, int EH) {
    int t = blockIdx.x * blockDim.x + threadIdx.x;
    if (t >= EH) return;
    int e = t >> 2, h = t & 3;
    int s = src[e], d = dst[e];
    size_t dh = (size_t)d * HEADS + h;
    float m = funmap(mhat[dh]);
    float p = expf(elog[t] - m);
    atomicAdd(z + dh, p);

    float ef[EDIM];
    const float* efp = edge_fea + (size_t)e * EDIM;
    #pragma unroll
    for (int i = 0; i < EDIM; ++i) ef[i] = efp[i];
    float* fdf = ftf + dh * EDIM;
    #pragma unroll
    for (int j = 0; j < EDIM; ++j) {
        float epv = bedg[h * EDIM + j];
        #pragma unroll
        for (int i = 0; i < EDIM; ++i)
            epv += ef[i] * Wedg[i * (HEADS * EDIM) + h * EDIM + j];
        atomicAdd(fdf + j, p * epv);
    }
    const float* fsrc = featp + (size_t)s * FIN + h * DOUT;
    float* fdst = ft + dh * DOUT;
    #pragma unroll
    for (int dd = 0; dd < DOUT; ++dd)
        atomicAdd(fdst + dd, p * fsrc[dd]);
}

// ---------------------------------------------------------------------------
// Kernel 5: out = ([ft_f | ft] / z) @ W_out + b_out + bias, via fp32 WMMA.
// Rows = (n,h) pairs (R = N*H); K = 21 padded to 24; one 16-col tile.
// No EXEC-divergent loads; bias folds to a 4-entry table indexed by r&3.
// ---------------------------------------------------------------------------
__global__ __launch_bounds__(128)
void k_out_proj(const float* __restrict__ ftf, const float* __restrict__ ft,
                const float* __restrict__ z, const float* __restrict__ Wout,
                const float* __restrict__ bout, const float* __restrict__ bias,
                float* __restrict__ out, int R) {
    __shared__ float sW[24 * 16];   // W_out zero-padded 21 -> 24 rows
    for (int i = threadIdx.x; i < 24 * 16; i += 128) {
        int k = i >> 4;
        sW[i] = (k < (EDIM + DOUT)) ? Wout[i] : 0.0f;
    }
    __syncthreads();

    const int lane  = threadIdx.x & 31;
    const int wave  = threadIdx.x >> 5;
    const int n16   = lane & 15;
    const int half  = lane >> 4;
    const int kbase = half * 2;
    const int rowBase = blockIdx.x * 64 + wave * 16;
    int arow = rowBase + n16;
    int crow = (arow < R) ? arow : (R - 1);     // clamp: dead rows never stored

    float zr = z[crow];
    float invz = (zr > 0.0f) ? (1.0f / zr) : 0.0f;
    const float* pf = ftf + (size_t)crow * EDIM;
    const float* pt = ft  + (size_t)crow * DOUT;

    v8f acc = {};
    #pragma unroll
    for (int kk = 0; kk < 24; kk += 4) {
        int c0 = kk + kbase, c1 = c0 + 1;
        v2f a, b;
        float a0 = (c0 < EDIM) ? pf[c0] : ((c0 < EDIM + DOUT) ? pt[c0 - EDIM] : 0.0f);
        float a1 = (c1 < EDIM) ? pf[c1] : ((c1 < EDIM + DOUT) ? pt[c1 - EDIM] : 0.0f);
        a.x = a0 * invz; a.y = a1 * invz;
        b.x = sW[c0 * 16 + n16];
        b.y = sW[c1 * 16 + n16];
        acc = __builtin_amdgcn_wmma_f32_16x16x4_f32(
            false, a, false, b, (short)0, acc, false, false);
    }

    // bias table: rows are (n,h) pairs; rowBase%16==0 and half*8%4==0 so the
    // head of output row (rowBase + r + 8*half) is just r&3.
    float badd[4];
    #pragma unroll
    for (int h = 0; h < 4; ++h) badd[h] = bout[n16] + bias[h * 16 + n16];

    if (rowBase + 16 <= R) {
        float* outp = out + (size_t)(rowBase + half * 8) * DOUT + n16;
        #pragma unroll
        for (int r = 0; r < 8; ++r)
            outp[(size_t)r * DOUT] = acc[r] + badd[r & 3];
    } else {
        #pragma unroll
        for (int r = 0; r < 8; ++r) {
            int drow = rowBase + r + half * 8;
            if (drow < R)
                out[(size_t)drow * DOUT + n16] = acc[r] + badd[r & 3];
        }
    }
}

// ---------------------------------------------------------------------------
extern "C" void kernel_launch(void* const* d_in, const int* in_sizes, int n_in,
                              void* d_out, int out_size, void* d_ws, size_t ws_size,
                              hipStream_t stream) {
    const float* feat     = (const float*)d_in[0];
    const float* edge_fea = (const float*)d_in[1];
    const int*   src      = (const int*)d_in[2];
    const int*   dst      = (const int*)d_in[3];
    const float* Wfc      = (const float*)d_in[4];
    const float* Wedg     = (const float*)d_in[5];
    const float* bedg     = (const float*)d_in[6];
    const float* attn_l   = (const float*)d_in[7];
    const float* attn_r   = (const float*)d_in[8];
    const float* attn_edg = (const float*)d_in[9];
    const float* Wout     = (const float*)d_in[10];
    const float* bout     = (const float*)d_in[11];
    const float* biasp    = (const float*)d_in[12];

    const int N  = in_sizes[0] / FIN;    // 100000
    const int E  = in_sizes[2];          // 1600000
    const int NH = N * HEADS;
    const int EH = E * HEADS;

    // workspace carve (floats). Atomic accumulators laid out contiguously so
    // one zero kernel covers mhat|z|ftf|ft.
    float* ws = (float*)d_ws;
    size_t off = 0;
    float*    featp = ws + off; off += (size_t)N * FIN;          // [N,64]
    float*    el    = ws + off; off += (size_t)NH;               // [N,H]
    float*    er    = ws + off; off += (size_t)NH;               // [N,H]
    float*    elog  = ws + off; off += (size_t)EH;               // [E,H]
    float*    zbase = ws + off;
    unsigned* mhat  = (unsigned*)(ws + off); off += (size_t)NH;  // [N,H] mapped max
    float*    zbuf  = ws + off; off += (size_t)NH;               // [N,H]
    float*    ftf   = ws + off; off += (size_t)NH * EDIM;        // [N,H,5]
    float*    ftb   = ws + off; off += (size_t)NH * DOUT;        // [N,H,16]

    int nzero = NH * (1 + 1 + EDIM + DOUT);
    k_zero<<<(nzero + 255) / 256, 256, 0, stream>>>(zbase, nzero);

    k_node_proj<<<(N + 63) / 64, 128, 0, stream>>>(feat, Wfc, featp, N);

    k_node_attn<<<(NH + 255) / 256, 256, 0, stream>>>(featp, attn_l, attn_r, el, er, NH);

    k_edge_logit<<<(EH + 255) / 256, 256, 0, stream>>>(
        edge_fea, src, dst, Wedg, bedg, attn_edg, el, er, elog, mhat, EH);

    k_edge_aggr<<<(EH + 255) / 256, 256, 0, stream>>>(
        edge_fea, src, dst, Wedg, bedg, featp, elog, mhat, zbuf, ftf, ftb, EH);

    const int R = NH;                    // 400000 rows, 64 | R
    k_out_proj<<<(R + 63) / 64, 128, 0, stream>>>(
        ftf, ftb, zbuf, Wout, bout, biasp, (float*)d_out, R);
}